// ResTemplateNet_48206712930685
// MI455X (gfx1250) — compile-verified
//
#include <hip/hip_runtime.h>
#include <hip/hip_bf16.h>

#define NODES 68000
#define FIN 256
#define CH 128
#define NE 2176000
#define NB 4
#define NPROT 17000
#define POOL_CHUNK 500   // 34 * 500 = 17000

typedef float v2f __attribute__((ext_vector_type(2)));
typedef float v8f __attribute__((ext_vector_type(8)));

__device__ __forceinline__ void atomAddF(float* p, float v) {
#if __has_builtin(__builtin_amdgcn_global_atomic_fadd_f32)
    // guaranteed hardware global_atomic_add_f32, no return
    __builtin_amdgcn_global_atomic_fadd_f32(p, v);
#else
    // relaxed device-scope fadd; lowers to global_atomic_add_f32 on gfx1250
    __hip_atomic_fetch_add(p, v, __ATOMIC_RELAXED, __HIP_MEMORY_SCOPE_AGENT);
#endif
}

__global__ void k_fill(float* __restrict__ p, float v, int n) {
    int i = blockIdx.x * blockDim.x + threadIdx.x;
    if (i < n) p[i] = v;
}

__global__ void k_degree(const int* __restrict__ dst, float* __restrict__ deg) {
    int e = blockIdx.x * blockDim.x + threadIdx.x;
    if (e < NE) atomAddF(&deg[dst[e]], 1.0f);
}

__global__ void k_rsqrt_inplace(float* __restrict__ d, int n) {
    int i = blockIdx.x * blockDim.x + threadIdx.x;
    if (i < n) d[i] = rsqrtf(d[i]);
}

// H[M,CH] = act(X)[M,K] @ W[K,CH] using V_WMMA_F32_16X16X4_F32,
// where act = ReLU when RELU_IN (layers 2/3 read the pre-activation buffer).
// Block = 256 threads = 8 waves; wave w owns output tile (blockIdx.x*16, w*16).
// A 16x4 f32 layout: lanes 0-15 -> M, VGPR0/1 = K=(0,1) for lanes 0-15, K=(2,3)
// for lanes 16-31. B 4x16 mirrored: lane%16 -> N, lane group selects K pair.
// D: VGPR r -> row (r + 8*(lane>=16)), col lane%16.
template<int K, bool RELU_IN>
__global__ void k_gemm_wmma(const float* __restrict__ X, const float* __restrict__ W,
                            float* __restrict__ H) {
    const int wave = threadIdx.x >> 5;
    const int lane = threadIdx.x & 31;
    const int l = lane & 15;
    const int g = lane >> 4;
    const size_t m0 = (size_t)blockIdx.x * 16;
    const int n0 = wave * 16;

    const float* xp = X + (m0 + l) * K + 2 * g;           // this lane's A row/K-pair
    const float* wp = W + (size_t)(2 * g) * CH + n0 + l;  // this lane's B col/K-pair

    v8f acc = 0.0f;
#pragma unroll 4
    for (int k = 0; k < K; k += 4) {
        v2f a = *(const v2f*)(xp + k);                    // A[m, k+2g .. k+2g+1]
        if (RELU_IN) {
            a.x = fmaxf(a.x, 0.0f);
            a.y = fmaxf(a.y, 0.0f);
        }
        v2f b;
        b.x = wp[(size_t)k * CH];                         // W[k+2g,   n]
        b.y = wp[(size_t)k * CH + CH];                    // W[k+2g+1, n]
        acc = __builtin_amdgcn_wmma_f32_16x16x4_f32(false, a, false, b,
                                                    (short)0, acc, false, false);
    }

    float* out = H + (m0 + 8 * g) * CH + n0 + l;
#pragma unroll
    for (int r = 0; r < 8; ++r) out[(size_t)r * CH] = acc[r];
}

// agg = h * dinv^2 + bias   (self-loop term + bias, seeds the scatter target)
__global__ void k_init_agg(const float* __restrict__ H, const float* __restrict__ dinv,
                           const float* __restrict__ bias, float* __restrict__ A) {
    int i = blockIdx.x * blockDim.x + threadIdx.x;
    if (i >= NODES * CH) return;
    int node = i >> 7;
    int c = i & 127;
    float di = dinv[node];
    A[i] = H[i] * di * di + bias[c];
}

// agg[dst] += dinv[src]*dinv[dst] * h[src] ; 32 lanes/edge, 4 channels/lane.
// h (34.8 MB) is L2-resident, so both the b128 gather and the f32 atomics
// are served out of the 192 MB L2.
__global__ void k_edges(const int* __restrict__ src, const int* __restrict__ dst,
                        const float* __restrict__ dinv, const float* __restrict__ H,
                        float* __restrict__ A) {
    int gid = blockIdx.x * blockDim.x + threadIdx.x;
    int e = gid >> 5;
    if (e >= NE) return;
    int q = (gid & 31) << 2;
    int s = src[e];
    int d = dst[e];
    float coef = dinv[s] * dinv[d];
    const float4 hv = *(const float4*)(H + (size_t)s * CH + q);
    float* ap = A + (size_t)d * CH + q;
    atomAddF(ap + 0, coef * hv.x);
    atomAddF(ap + 1, coef * hv.y);
    atomAddF(ap + 2, coef * hv.z);
    atomAddF(ap + 3, coef * hv.w);
}

// pooled[b,c] += sum_n mask[b,n] * relu(agg[(b*NPROT+n)*CH + c])
__global__ void k_pool(const float* __restrict__ A, const int* __restrict__ mask,
                       float* __restrict__ pooled) {
    int c = threadIdx.x;             // 0..127
    int b = blockIdx.y;              // 0..3
    int n0 = blockIdx.x * POOL_CHUNK;
    float sum = 0.0f;
    for (int n = n0; n < n0 + POOL_CHUNK; ++n) {
        if (mask[b * NPROT + n]) {   // uniform (scalar) branch across the block
            sum += fmaxf(A[((size_t)b * NPROT + n) * CH + c], 0.0f);
        }
    }
    atomAddF(&pooled[b * CH + c], sum);
}

// tiny MLP: [4,128] -> 256 -> 64 -> 16 -> 1, single block
__global__ void k_mlp(const float* __restrict__ pooled,
                      const float* __restrict__ lw1, const float* __restrict__ lb1,
                      const float* __restrict__ lw2, const float* __restrict__ lb2,
                      const float* __restrict__ lw3, const float* __restrict__ lb3,
                      const float* __restrict__ lw4, const float* __restrict__ lb4,
                      float* __restrict__ out) {
    __shared__ float sp[NB * CH];
    __shared__ float s1[NB * 256];
    __shared__ float s2[NB * 64];
    __shared__ float s3[NB * 16];
    int t = threadIdx.x;  // 0..255
    for (int i = t; i < NB * CH; i += 256) sp[i] = pooled[i];
    __syncthreads();
    {
        float bias = lb1[t];
        for (int b = 0; b < NB; ++b) {
            float acc = bias;
            for (int k = 0; k < CH; ++k) acc += sp[b * CH + k] * lw1[k * 256 + t];
            s1[b * 256 + t] = fmaxf(acc, 0.0f);
        }
    }
    __syncthreads();
    if (t < 64) {
        float bias = lb2[t];
        for (int b = 0; b < NB; ++b) {
            float acc = bias;
            for (int k = 0; k < 256; ++k) acc += s1[b * 256 + k] * lw2[k * 64 + t];
            s2[b * 64 + t] = fmaxf(acc, 0.0f);
        }
    }
    __syncthreads();
    if (t < 16) {
        float bias = lb3[t];
        for (int b = 0; b < NB; ++b) {
            float acc = bias;
            for (int k = 0; k < 64; ++k) acc += s2[b * 64 + k] * lw3[k * 16 + t];
            s3[b * 16 + t] = fmaxf(acc, 0.0f);
        }
    }
    __syncthreads();
    if (t < NB) {
        float acc = lb4[0];
        for (int k = 0; k < 16; ++k) acc += s3[t * 16 + k] * lw4[k];
        out[t] = acc;
    }
}

extern "C" void kernel_launch(void* const* d_in, const int* in_sizes, int n_in,
                              void* d_out, int out_size, void* d_ws, size_t ws_size,
                              hipStream_t stream) {
    (void)in_sizes; (void)n_in; (void)out_size; (void)ws_size;

    const float* x   = (const float*)d_in[0];
    const int*   ei  = (const int*)d_in[1];
    const int*   msk = (const int*)d_in[2];
    const float* W1  = (const float*)d_in[3];
    const float* b1  = (const float*)d_in[4];
    const float* W2  = (const float*)d_in[5];
    const float* b2  = (const float*)d_in[6];
    const float* W3  = (const float*)d_in[7];
    const float* b3  = (const float*)d_in[8];
    const float* lw1 = (const float*)d_in[9];
    const float* lb1 = (const float*)d_in[10];
    const float* lw2 = (const float*)d_in[11];
    const float* lb2 = (const float*)d_in[12];
    const float* lw3 = (const float*)d_in[13];
    const float* lb3 = (const float*)d_in[14];
    const float* lw4 = (const float*)d_in[15];
    const float* lb4 = (const float*)d_in[16];

    const int* src = ei;        // edge_index[0]
    const int* dst = ei + NE;   // edge_index[1]
    float* out = (float*)d_out;

    // workspace layout (floats): dinv | H | A | pooled  (~70 MB)
    float* ws     = (float*)d_ws;
    float* dinv   = ws;                               // NODES (padded to 68096)
    float* H      = ws + 68096;                       // NODES*CH (GEMM output)
    float* A      = H + (size_t)NODES * CH;           // NODES*CH (agg, pre-ReLU)
    float* pooled = A + (size_t)NODES * CH;           // NB*CH

    const int elems = NODES * CH;

    // degrees (self-loop = 1) -> dinv, zero pooled accumulator
    k_fill<<<(NODES + 255) / 256, 256, 0, stream>>>(dinv, 1.0f, NODES);
    k_degree<<<(NE + 255) / 256, 256, 0, stream>>>(dst, dinv);
    k_rsqrt_inplace<<<(NODES + 255) / 256, 256, 0, stream>>>(dinv, NODES);
    k_fill<<<(NB * CH + 255) / 256, 256, 0, stream>>>(pooled, 0.0f, NB * CH);

    // ---- layer 1 (input x, K=256, no input ReLU) ----
    k_gemm_wmma<FIN, false><<<NODES / 16, 256, 0, stream>>>(x, W1, H);
    k_init_agg<<<(elems + 255) / 256, 256, 0, stream>>>(H, dinv, b1, A);
    k_edges<<<(NE * 32) / 256, 256, 0, stream>>>(src, dst, dinv, H, A);
    k_pool<<<dim3(NPROT / POOL_CHUNK, NB), CH, 0, stream>>>(A, msk, pooled);

    // ---- layer 2 (input relu(A), K=128) ----
    k_gemm_wmma<CH, true><<<NODES / 16, 256, 0, stream>>>(A, W2, H);
    k_init_agg<<<(elems + 255) / 256, 256, 0, stream>>>(H, dinv, b2, A);
    k_edges<<<(NE * 32) / 256, 256, 0, stream>>>(src, dst, dinv, H, A);
    k_pool<<<dim3(NPROT / POOL_CHUNK, NB), CH, 0, stream>>>(A, msk, pooled);

    // ---- layer 3 (input relu(A), K=128) ----
    k_gemm_wmma<CH, true><<<NODES / 16, 256, 0, stream>>>(A, W3, H);
    k_init_agg<<<(elems + 255) / 256, 256, 0, stream>>>(H, dinv, b3, A);
    k_edges<<<(NE * 32) / 256, 256, 0, stream>>>(src, dst, dinv, H, A);
    k_pool<<<dim3(NPROT / POOL_CHUNK, NB), CH, 0, stream>>>(A, msk, pooled);

    k_mlp<<<1, 256, 0, stream>>>(pooled, lw1, lb1, lw2, lb2, lw3, lb3, lw4, lb4, out);
}